// STCGAT_72619307041568
// MI455X (gfx1250) — compile-verified
//
#include <hip/hip_runtime.h>
#include <hip/hip_bf16.h>

// ---------------- model dims ----------------
#define NN    307   // nodes
#define FIN   64    // input dim
#define CC    64    // rnn units
#define TT    12    // seq len
#define HOR   12    // horizon
#define HH    4     // heads
#define GF    64    // per-head dim
#define EDIM  10    // embed dim
#define BB    16    // batch
#define DCAT  128   // FIN + CC
#define HGF   256   // HH*GF

typedef float v2f __attribute__((ext_vector_type(2)));
typedef float v8f __attribute__((ext_vector_type(8)));

__device__ __forceinline__ v8f wmma_f32_k4(v2f a, v2f b, v8f c) {
    // V_WMMA_F32_16X16X4_F32 : D = A(16x4) * B(4x16) + C(16x16), fp32 throughout
    return __builtin_amdgcn_wmma_f32_16x16x4_f32(false, a, false, b, (short)0, c,
                                                 false, false);
}

__device__ __forceinline__ float apply_act(float val, int act) {
    if (act == 1)      val = fmaxf(val, 0.f);
    else if (act == 2) val = 1.f / (1.f + __expf(-val));
    else if (act == 3) val = tanhf(val);
    return val;
}

// =====================================================================
// fp32 WMMA GEMM, 16x64 tile per wave (4 accumulators share one A frag).
// D[M,Ncol] = act(Afrag @ B + bias + Cres)
// REQUIRES: M%16==0, Ncol%64==0, K%4==0.
// amode 0: A = Amat[M,K] row-major
// amode 1: A row = concat(x[b,t,n,:FIN], h[row,:CC])          (K==128)
// amode 2: A row = concat(x[b,t,n,:FIN], r[row]*h[row])       (K==128)
// act: 0=none 1=relu 2=sigmoid 3=tanh
// =====================================================================
__global__ __launch_bounds__(32) void gemm_wmma_n64_kernel(
    const float* __restrict__ Amat, const float* __restrict__ X,
    const float* __restrict__ Hst, const float* __restrict__ Zr,
    const float* __restrict__ B, const float* __restrict__ bias,
    const float* __restrict__ Cres, float* __restrict__ D,
    int M, int Ncol, int K, int act, int amode, int t)
{
    const int lane = threadIdx.x & 31;
    const int hl   = lane >> 4;     // half-wave: K offset 0 / +2
    const int l    = lane & 15;
    const int m0   = blockIdx.x * 16;
    const int n0   = blockIdx.y * 64;
    const int row  = m0 + l;

    v8f acc[4];
#pragma unroll
    for (int tj = 0; tj < 4; ++tj) acc[tj] = (v8f){0.f,0.f,0.f,0.f,0.f,0.f,0.f,0.f};

    const float* Bp = B + (size_t)(2 * hl) * Ncol + n0 + l;
    const size_t bstep = (size_t)4 * Ncol;

    auto kstep = [&](v2f a) {
#pragma unroll
        for (int tj = 0; tj < 4; ++tj) {
            v2f b = { Bp[tj * 16], Bp[Ncol + tj * 16] };
            acc[tj] = wmma_f32_k4(a, b, acc[tj]);
        }
        Bp += bstep;
    };

    if (amode == 0) {
        const float* Ap = Amat + (size_t)row * K + 2 * hl;
        for (int k = 0; k < K; k += 4) {
            v2f a = *(const v2f*)Ap; Ap += 4;
            kstep(a);
        }
    } else {
        const int bb = row / NN, n = row % NN;
        const float* Xp = X + ((size_t)(bb * TT + t) * NN + n) * FIN + 2 * hl;
        for (int k = 0; k < FIN; k += 4) {
            v2f a = *(const v2f*)Xp; Xp += 4;
            kstep(a);
        }
        const float* Hp = Hst + (size_t)row * CC + 2 * hl;
        if (amode == 1) {
            for (int k = 0; k < CC; k += 4) {
                v2f a = *(const v2f*)Hp; Hp += 4;
                kstep(a);
            }
        } else {
            const float* Zp = Zr + (size_t)row * DCAT + CC + 2 * hl;
            for (int k = 0; k < CC; k += 4) {
                v2f hv = *(const v2f*)Hp; Hp += 4;
                v2f rv = *(const v2f*)Zp; Zp += 4;
                v2f a = { hv[0] * rv[0], hv[1] * rv[1] };
                kstep(a);
            }
        }
    }

#pragma unroll
    for (int tj = 0; tj < 4; ++tj) {
        const int col = n0 + tj * 16 + l;
        const float bs = bias ? bias[col] : 0.f;
#pragma unroll
        for (int v = 0; v < 8; ++v) {
            const int r2 = m0 + v + 8 * hl;
            float val = acc[tj][v] + bs;
            if (Cres) val += Cres[(size_t)r2 * Ncol + col];
            D[(size_t)r2 * Ncol + col] = apply_act(val, act);
        }
    }
}

// =====================================================================
// Flexible fp32 WMMA GEMM, 16x16 tile per wave (for ragged N, e.g. N=16)
// REQUIRES: M%16==0, Ncol%16==0, K%4==0.
// =====================================================================
__global__ __launch_bounds__(32) void gemm_wmma_kernel(
    const float* __restrict__ A, const float* __restrict__ B,
    const float* __restrict__ bias, const float* __restrict__ Cres,
    float* __restrict__ D, int M, int Ncol, int K, int act)
{
    const int lane = threadIdx.x & 31;
    const int hl   = lane >> 4;
    const int l    = lane & 15;
    const int m0   = blockIdx.x * 16;
    const int n0   = blockIdx.y * 16;

    v8f acc = {0.f, 0.f, 0.f, 0.f, 0.f, 0.f, 0.f, 0.f};

    const float* Ap = A + (size_t)(m0 + l) * K + 2 * hl;
    const float* Bp = B + (size_t)(2 * hl) * Ncol + (n0 + l);
    const size_t bstep = (size_t)4 * Ncol;

    for (int k = 0; k < K; k += 4) {
        v2f a = *(const v2f*)(Ap);
        v2f b = { Bp[0], Bp[Ncol] };
        acc = wmma_f32_k4(a, b, acc);
        Ap += 4;
        Bp += bstep;
    }

    const int col = n0 + l;
    const float bs = bias ? bias[col] : 0.f;
#pragma unroll
    for (int v = 0; v < 8; ++v) {
        const int row = m0 + v + 8 * hl;
        float val = acc[v] + bs;
        if (Cres) val += Cres[(size_t)row * Ncol + col];
        D[(size_t)row * Ncol + col] = apply_act(val, act);
    }
}

// =====================================================================
// Fused GAT attention for one (b,h) and a 16-row tile:
//   es_i = hg_i . a_s ; ed_j = hg_j . a_d      (computed in-kernel)
//   e_ij = leaky_relu(es_i + ed_j, 0.2) + adjbias_ij
//   p    = softmax_j(e)                         (staged in LDS, stable)
//   out[i,f] = sum_j p_ij * hg[b,j,h,f]         via WMMA (4 waves x 16 cols)
// hg layout: [B,N,H,GF] with one extra zeroed row appended (row NN) so the
// padded K=308 inner loop needs no guards (P[.,307]==0 anyway).
// out layout: [B,N,H*GF]
// =====================================================================
#define NP   308          // K padded to multiple of 4
#define LSTR 312          // LDS row stride (skewed)

__global__ __launch_bounds__(128) void gat_attn_kernel(
    const float* __restrict__ hg, const float* __restrict__ a_s,
    const float* __restrict__ a_d, const float* __restrict__ adjb,
    float* __restrict__ outp)
{
    __shared__ float P[16 * LSTR];
    __shared__ float edS[NP];
    __shared__ float esS[16];
    __shared__ float red[16 * 8];
    __shared__ float rowmax[16];
    __shared__ float rowsum[16];

    const int bh = blockIdx.x;             // 0 .. B*H-1
    const int b  = bh / HH;
    const int h  = bh % HH;
    const int i0 = blockIdx.y * 16;
    const int tid = threadIdx.x;

    // e_dst for all j (cooperative), e_src for this tile's 16 rows
    for (int j = tid; j < NP; j += 128) {
        float s = 0.f;
        if (j < NN) {
            const float* rowp = hg + ((size_t)(b * NN + j) * HH + h) * GF;
            const float* ad = a_d + h * GF;
#pragma unroll 8
            for (int f = 0; f < GF; ++f) s += rowp[f] * ad[f];
        }
        edS[j] = s;
    }
    if (tid < 16) {
        const int i = i0 + tid;
        float s = 0.f;
        if (i < NN) {
            const float* rowp = hg + ((size_t)(b * NN + i) * HH + h) * GF;
            const float* as = a_s + h * GF;
#pragma unroll 8
            for (int f = 0; f < GF; ++f) s += rowp[f] * as[f];
        }
        esS[tid] = s;
    }
    __syncthreads();

    const int r = tid >> 3;                // local row 0..15
    const int c = tid & 7;                 // 8 threads per row
    const int i = i0 + r;
    const bool rv = (i < NN);
    const float esi = esS[r];

    // pass 1: raw logits + per-row max
    float m = -3.0e38f;
    for (int j = c; j < NP; j += 8) {
        float e = -3.0e38f;
        if (rv && j < NN) {
            float x = esi + edS[j];
            x = (x > 0.f) ? x : 0.2f * x;
            e = x + adjb[i * NN + j];
        }
        P[r * LSTR + j] = e;
        m = fmaxf(m, e);
    }
    red[r * 8 + c] = m;
    __syncthreads();
    if (c == 0) {
        float mm = red[r * 8];
#pragma unroll
        for (int k = 1; k < 8; ++k) mm = fmaxf(mm, red[r * 8 + k]);
        rowmax[r] = mm;
    }
    __syncthreads();

    // pass 2: exp + per-row sum (invalid cols -> 0 so they add nothing)
    const float mrow = rowmax[r];
    float s = 0.f;
    for (int j = c; j < NP; j += 8) {
        float p = 0.f;
        if (rv && j < NN) p = __expf(P[r * LSTR + j] - mrow);
        P[r * LSTR + j] = p;
        s += p;
    }
    red[r * 8 + c] = s;
    __syncthreads();
    if (c == 0) {
        float ss = 0.f;
#pragma unroll
        for (int k = 0; k < 8; ++k) ss += red[r * 8 + k];
        rowsum[r] = ss;
    }
    __syncthreads();

    // pass 3: out[16,64] = P[16,308] @ Hg[308,64] via WMMA, 4 waves x 16 cols
    const int wave = tid >> 5;
    const int lane = tid & 31;
    const int hl = lane >> 4;
    const int l  = lane & 15;
    const int col = wave * 16 + l;

    const float* Prow = P + l * LSTR + 2 * hl;
    const float* Hp = hg + ((size_t)(b * NN + 2 * hl) * HH + h) * GF + col;
    const size_t hstep = (size_t)4 * HH * GF;

    v8f acc = {0.f, 0.f, 0.f, 0.f, 0.f, 0.f, 0.f, 0.f};
    for (int k = 0; k < NP; k += 4) {
        v2f a = *(const v2f*)(Prow);        // ds_load_b64
        v2f bb = { Hp[0], Hp[HH * GF] };
        acc = wmma_f32_k4(a, bb, acc);
        Prow += 4;
        Hp += hstep;
    }

#pragma unroll
    for (int v = 0; v < 8; ++v) {
        const int lr = v + 8 * hl;
        const int row = i0 + lr;
        if (row < NN) {
            const float inv = 1.f / rowsum[lr];
            outp[(size_t)(b * NN + row) * HGF + h * GF + col] = acc[v] * inv;
        }
    }
}

// ---------------- small VALU kernels ----------------
__global__ void fill_kernel(float* p, int n, float v) {
    int i = blockIdx.x * blockDim.x + threadIdx.x;
    if (i < n) p[i] = v;
}

// source [B,N,F,T] -> xF [B,T,N,F] and time-flipped xB
__global__ void transpose_x_kernel(const float* __restrict__ src,
                                   float* __restrict__ xF, float* __restrict__ xB) {
    int idx = blockIdx.x * blockDim.x + threadIdx.x;
    const int total = BB * TT * NN * FIN;
    if (idx >= total) return;
    int f = idx % FIN; int rest = idx / FIN;
    int n = rest % NN; rest /= NN;
    int t = rest % TT; int b = rest / TT;
    float v = src[((b * NN + n) * FIN + f) * TT + t];
    xF[idx] = v;
    xB[((b * TT + (TT - 1 - t)) * NN + n) * FIN + f] = v;
}

// adjacency bias: (emb_i . emb_j) > 0 ? 0 : -1e9
__global__ void bias_kernel(const float* __restrict__ emb, float* __restrict__ adjb) {
    int idx = blockIdx.x * blockDim.x + threadIdx.x;
    if (idx >= NN * NN) return;
    int i = idx / NN, j = idx % NN;
    float s = 0.f;
#pragma unroll
    for (int e = 0; e < EDIM; ++e) s += emb[i * EDIM + e] * emb[j * EDIM + e];
    adjb[idx] = (s > 0.f) ? 0.f : -1e9f;
}

// Wh [H,128,GF] -> packed [128, H*GF]
__global__ void pack_wh_kernel(const float* __restrict__ Wh, float* __restrict__ Wp) {
    int idx = blockIdx.x * blockDim.x + threadIdx.x;
    if (idx >= DCAT * HGF) return;
    int d = idx / HGF, rem = idx % HGF;
    int h = rem / GF, f = rem % GF;
    Wp[idx] = Wh[(h * DCAT + d) * GF + f];
}

// W [Cout,K] -> Wp [K,Cout]
__global__ void pack_tr_kernel(const float* __restrict__ W, float* __restrict__ Wp,
                               int Cout, int K) {
    int idx = blockIdx.x * blockDim.x + threadIdx.x;
    if (idx >= Cout * K) return;
    int co = idx / K, k = idx % K;
    Wp[k * Cout + co] = W[co * K + k];
}

// p_w2 [768,12] -> [768,16] zero-padded ; p_b2 [12] -> [16]
__global__ void pack_pw2_kernel(const float* __restrict__ w, const float* __restrict__ b,
                                float* __restrict__ wp, float* __restrict__ bp) {
    int idx = blockIdx.x * blockDim.x + threadIdx.x;
    if (idx >= (TT * CC) * 16) return;
    int k = idx / 16, c = idx % 16;
    wp[idx] = (c < HOR) ? w[k * HOR + c] : 0.f;
    if (idx < 16) bp[idx] = (idx < HOR) ? b[idx] : 0.f;
}

// outPad [4912,16] -> d_out [4912,12]
__global__ void copy_out_kernel(const float* __restrict__ yp, float* __restrict__ out) {
    int idx = blockIdx.x * blockDim.x + threadIdx.x;
    if (idx >= BB * NN * HOR) return;
    int r = idx / HOR, c = idx % HOR;
    out[idx] = yp[r * 16 + c];
}

// hn = z*h + (1-z)*hc ; write state[t]
__global__ void gru_update_kernel(const float* __restrict__ zr,
                                  const float* __restrict__ hc,
                                  float* __restrict__ hst, float* __restrict__ states, int t) {
    int idx = blockIdx.x * blockDim.x + threadIdx.x;
    if (idx >= BB * NN * CC) return;
    int c = idx % CC; int row = idx / CC;
    int n = row % NN; int b = row / NN;
    float z = zr[row * DCAT + c];
    float hn = z * hst[idx] + (1.f - z) * hc[idx];
    hst[idx] = hn;
    states[((b * TT + t) * NN + n) * CC + c] = hn;
}

// temporal conv over T, kernel 3, pad 1 : tc[b,t,n,co]
__global__ void tconv_kernel(const float* __restrict__ st, const float* __restrict__ w,
                             const float* __restrict__ bvec, float* __restrict__ tc) {
    int idx = blockIdx.x * blockDim.x + threadIdx.x;
    if (idx >= BB * TT * NN * CC) return;
    int co = idx % CC; int rest = idx / CC;
    int n = rest % NN; rest /= NN;
    int t = rest % TT; int b = rest / TT;
    float acc = bvec[co];
#pragma unroll
    for (int kt = 0; kt < 3; ++kt) {
        int tt = t + kt - 1;
        if (tt < 0 || tt >= TT) continue;
        const float* sp = st + ((b * TT + tt) * NN + n) * CC;
        const float* wp = w + co * CC * 3 + kt;
        for (int ci = 0; ci < CC; ++ci) acc += sp[ci] * wp[ci * 3];
    }
    tc[idx] = acc;
}

// layernorm over channel dim; writes into hout half (0: cols 0..63, 1: 64..127)
__global__ void ln_kernel(const float* __restrict__ y, const float* __restrict__ g,
                          const float* __restrict__ be, float* __restrict__ hout, int halfsel) {
    int row = blockIdx.x * blockDim.x + threadIdx.x;
    if (row >= BB * TT * NN) return;
    const float* yr = y + (size_t)row * CC;
    float m = 0.f;
    for (int c2 = 0; c2 < CC; ++c2) m += yr[c2];
    m *= (1.f / CC);
    float v = 0.f;
    for (int c2 = 0; c2 < CC; ++c2) { float d = yr[c2] - m; v += d * d; }
    v *= (1.f / CC);
    float inv = rsqrtf(v + 1e-5f);
    float* o = hout + (size_t)row * (2 * CC) + halfsel * CC;
    for (int c2 = 0; c2 < CC; ++c2) o[c2] = (yr[c2] - m) * inv * g[c2] + be[c2];
}

// causal dilated conv, kernel 2: y[t] = relu(b + w0*x[t-d] + w1*x[t])
__global__ void tcn_kernel(const float* __restrict__ in, const float* __restrict__ w,
                           const float* __restrict__ bvec, float* __restrict__ out,
                           int cin, int dil) {
    int idx = blockIdx.x * blockDim.x + threadIdx.x;
    if (idx >= BB * TT * NN * CC) return;
    int co = idx % CC; int rest = idx / CC;
    int n = rest % NN; rest /= NN;
    int t = rest % TT; int b = rest / TT;
    float acc = bvec[co];
    const float* in1 = in + (size_t)((b * TT + t) * NN + n) * cin;
    const float* in0 = (t - dil >= 0) ? in + (size_t)((b * TT + (t - dil)) * NN + n) * cin
                                      : nullptr;
    const float* wr = w + (size_t)co * cin * 2;
    for (int ci = 0; ci < cin; ++ci) {
        acc += wr[ci * 2 + 1] * in1[ci];
        if (in0) acc += wr[ci * 2] * in0[ci];
    }
    out[idx] = fmaxf(acc, 0.f);
}

// [B,T,N,64] -> xf [B,N,T*64]
__global__ void to_xf_kernel(const float* __restrict__ y, float* __restrict__ xf) {
    int idx = blockIdx.x * blockDim.x + threadIdx.x;
    if (idx >= BB * TT * NN * CC) return;
    int c = idx % CC; int rest = idx / CC;
    int n = rest % NN; rest /= NN;
    int t = rest % TT; int b = rest / TT;
    xf[((b * NN + n) * TT + t) * CC + c] = y[idx];
}

// =====================================================================
// Host orchestration
// =====================================================================
static inline int divup(int a, int b) { return (a + b - 1) / b; }

extern "C" void kernel_launch(void* const* d_in, const int* in_sizes, int n_in,
                              void* d_out, int out_size, void* d_ws, size_t ws_size,
                              hipStream_t stream) {
    (void)in_sizes; (void)n_in; (void)out_size; (void)ws_size;
    auto F = [&](int i) { return (const float*)d_in[i]; };

    // setup_inputs() insertion order
    const float* src      = F(0);
    const float* node_emb = F(1);
    struct EncP {
        const float *zrWh, *zrAs, *zrAd, *zrWo, *zrB;
        const float *hhWh, *hhAs, *hhAd, *hhWo, *hhB;
        const float *tcw, *tcb, *rcw, *rcb, *lng, *lnb;
    };
    auto load_enc = [&](int base) {
        EncP e;
        e.zrWh = F(base + 0);  e.zrAs = F(base + 1);  e.zrAd = F(base + 2);
        e.zrWo = F(base + 3);  e.zrB  = F(base + 4);
        e.hhWh = F(base + 5);  e.hhAs = F(base + 6);  e.hhAd = F(base + 7);
        e.hhWo = F(base + 8);  e.hhB  = F(base + 9);
        e.tcw  = F(base + 10); e.tcb  = F(base + 11);
        e.rcw  = F(base + 12); e.rcb  = F(base + 13);
        e.lng  = F(base + 14); e.lnb  = F(base + 15);
        return e;
    };
    EncP encF = load_enc(2);
    EncP encB = load_enc(18);
    const float* tcnW[3] = { F(34), F(35), F(36) };
    const float* tcnB[3] = { F(37), F(38), F(39) };
    const float* downW = F(40); const float* downB = F(41);
    const float* pW1 = F(42);   const float* pB1 = F(43);
    const float* pW2 = F(44);   const float* pB2 = F(45);

    // -------- workspace carve-out --------
    float* ws = (float*)d_ws;
    size_t off = 0;
    auto alloc = [&](size_t n) { float* p = ws + off; off += (n + 63) & ~(size_t)63; return p; };
    const size_t BTNF = (size_t)BB * TT * NN * FIN;          // 3,772,416
    float* xF     = alloc(BTNF);
    float* xB     = alloc(BTNF);
    float* adjb   = alloc((size_t)NN * NN);
    float* WpZRf  = alloc(DCAT * HGF);
    float* WpHHf  = alloc(DCAT * HGF);
    float* WpZRb  = alloc(DCAT * HGF);
    float* WpHHb  = alloc(DCAT * HGF);
    float* rconvPf= alloc(FIN * CC);
    float* rconvPb= alloc(FIN * CC);
    float* downP  = alloc(2 * CC * CC);
    float* pW2p   = alloc((size_t)TT * CC * 16);
    float* pB2p   = alloc(16);
    float* hstate = alloc((size_t)BB * NN * CC);
    float* hg     = alloc((size_t)BB * NN * HGF + HGF);      // +1 padded row
    float* attn   = alloc((size_t)BB * NN * HGF);
    float* zrb    = alloc((size_t)BB * NN * DCAT);
    float* hcb    = alloc((size_t)BB * NN * CC);
    float* statesF= alloc(BTNF);
    float* statesB= alloc(BTNF);
    float* tcb_   = alloc(BTNF);
    float* ytmp   = alloc(BTNF);
    float* hout   = alloc(2 * BTNF);
    float* y1b    = alloc(BTNF);
    float* y2b    = alloc(BTNF);
    float* yfin   = alloc(BTNF);
    float* xfb    = alloc(BTNF);
    float* h1b    = alloc((size_t)BB * NN * TT * CC);
    float* outPad = alloc((size_t)BB * NN * 16);

    const int EW = 256;
    auto g1 = [&](size_t n) { return dim3((unsigned)divup((int)n, EW)); };
    // wide-tile GEMM (Ncol % 64 == 0), with fused A-concat modes
    auto gemm64 = [&](const float* A, const float* X, const float* Hst, const float* Zr,
                      const float* Bm, const float* bias, const float* Cres,
                      float* D, int M, int Ncol, int K, int act, int amode, int t) {
        dim3 grid(divup(M, 16), divup(Ncol, 64));
        gemm_wmma_n64_kernel<<<grid, 32, 0, stream>>>(A, X, Hst, Zr, Bm, bias, Cres, D,
                                                      M, Ncol, K, act, amode, t);
    };
    auto gemm16 = [&](const float* A, const float* Bm, const float* bias, const float* Cres,
                      float* D, int M, int Ncol, int K, int act) {
        dim3 grid(divup(M, 16), divup(Ncol, 16));
        gemm_wmma_kernel<<<grid, 32, 0, stream>>>(A, Bm, bias, Cres, D, M, Ncol, K, act);
    };

    // -------- preprocessing --------
    transpose_x_kernel<<<g1(BTNF), EW, 0, stream>>>(src, xF, xB);
    bias_kernel<<<g1(NN * NN), EW, 0, stream>>>(node_emb, adjb);
    pack_wh_kernel<<<g1(DCAT * HGF), EW, 0, stream>>>(encF.zrWh, WpZRf);
    pack_wh_kernel<<<g1(DCAT * HGF), EW, 0, stream>>>(encF.hhWh, WpHHf);
    pack_wh_kernel<<<g1(DCAT * HGF), EW, 0, stream>>>(encB.zrWh, WpZRb);
    pack_wh_kernel<<<g1(DCAT * HGF), EW, 0, stream>>>(encB.hhWh, WpHHb);
    pack_tr_kernel<<<g1(CC * FIN), EW, 0, stream>>>(encF.rcw, rconvPf, CC, FIN);
    pack_tr_kernel<<<g1(CC * FIN), EW, 0, stream>>>(encB.rcw, rconvPb, CC, FIN);
    pack_tr_kernel<<<g1(CC * 2 * CC), EW, 0, stream>>>(downW, downP, CC, 2 * CC);
    pack_pw2_kernel<<<g1((size_t)TT * CC * 16), EW, 0, stream>>>(pW2, pB2, pW2p, pB2p);
    // zero the padded hg row so the guard-free attention K-loop reads finite zeros
    fill_kernel<<<g1(HGF), EW, 0, stream>>>(hg + (size_t)BB * NN * HGF, HGF, 0.f);

    const int MBN  = BB * NN;        // 4912  (= 307*16)
    const int MBTN = BB * TT * NN;   // 58944 (= 3684*16)
    const dim3 agrid(BB * HH, divup(NN, 16));

    auto run_encoder = [&](const float* xenc, const EncP& E, const float* WpZR,
                           const float* WpHH, const float* rconvP,
                           float* states, int halfsel) {
        fill_kernel<<<g1(MBN * CC), EW, 0, stream>>>(hstate, MBN * CC, 0.f);
        for (int t = 0; t < TT; ++t) {
            // ---- zr gate : sigmoid(GAT(concat(x,h))) ----
            gemm64(nullptr, xenc, hstate, nullptr, WpZR, nullptr, nullptr,
                   hg, MBN, HGF, DCAT, 0, /*amode=*/1, t);
            gat_attn_kernel<<<agrid, 128, 0, stream>>>(hg, E.zrAs, E.zrAd, adjb, attn);
            gemm64(attn, nullptr, nullptr, nullptr, E.zrWo, E.zrB, nullptr,
                   zrb, MBN, DCAT, HGF, /*sigmoid*/2, 0, 0);
            // ---- candidate : tanh(GAT(concat(x, r*h))) ----
            gemm64(nullptr, xenc, hstate, zrb, WpHH, nullptr, nullptr,
                   hg, MBN, HGF, DCAT, 0, /*amode=*/2, t);
            gat_attn_kernel<<<agrid, 128, 0, stream>>>(hg, E.hhAs, E.hhAd, adjb, attn);
            gemm64(attn, nullptr, nullptr, nullptr, E.hhWo, E.hhB, nullptr,
                   hcb, MBN, CC, HGF, /*tanh*/3, 0, 0);
            gru_update_kernel<<<g1(MBN * CC), EW, 0, stream>>>(zrb, hcb, hstate, states, t);
        }
        // residual conv + temporal conv + relu + layernorm -> hout half
        tconv_kernel<<<g1((size_t)MBTN * CC), EW, 0, stream>>>(states, E.tcw, E.tcb, tcb_);
        gemm64(xenc, nullptr, nullptr, nullptr, rconvP, E.rcb, tcb_,
               ytmp, MBTN, CC, FIN, /*relu*/1, 0, 0);
        ln_kernel<<<g1(MBTN), EW, 0, stream>>>(ytmp, E.lng, E.lnb, hout, halfsel);
    };

    run_encoder(xF, encF, WpZRf, WpHHf, rconvPf, statesF, 0);
    run_encoder(xB, encB, WpZRb, WpHHb, rconvPb, statesB, 1);

    // -------- TCN: 3 causal dilated convs + downsample residual --------
    tcn_kernel<<<g1((size_t)MBTN * CC), EW, 0, stream>>>(hout, tcnW[0], tcnB[0], y1b, 2 * CC, 1);
    tcn_kernel<<<g1((size_t)MBTN * CC), EW, 0, stream>>>(y1b,  tcnW[1], tcnB[1], y2b, CC, 2);
    tcn_kernel<<<g1((size_t)MBTN * CC), EW, 0, stream>>>(y2b,  tcnW[2], tcnB[2], y1b, CC, 4);
    gemm64(hout, nullptr, nullptr, nullptr, downP, downB, y1b,
           yfin, MBTN, CC, 2 * CC, /*relu*/1, 0, 0);           // relu(tcn + down)

    // -------- final projection --------
    to_xf_kernel<<<g1((size_t)MBTN * CC), EW, 0, stream>>>(yfin, xfb);
    gemm64(xfb, nullptr, nullptr, nullptr, pW1, pB1, nullptr,
           h1b, MBN, HOR * GF, TT * CC, /*relu*/1, 0, 0);
    gemm16(h1b, pW2p, pB2p, nullptr, outPad, MBN, 16, HOR * GF, 0);
    copy_out_kernel<<<g1(MBN * HOR), EW, 0, stream>>>(outPad, (float*)d_out);
}